// CrossMotorFeatureExtractor_63797444215021
// MI455X (gfx1250) — compile-verified
//
#include <hip/hip_runtime.h>
#include <math.h>

typedef __attribute__((ext_vector_type(2))) float v2f;
typedef __attribute__((ext_vector_type(8))) float v8f;

#define CMFE_EPS 1e-8f

// DPP quad_perm cross-lane read (groups of 4 lanes), pure VALU — no LDS pipe.
// ctrl = sel0 | sel1<<2 | sel2<<4 | sel3<<6
template <int CTRL>
__device__ __forceinline__ float quad_perm_f32(float v) {
  int s = __builtin_bit_cast(int, v);
  int d = __builtin_amdgcn_update_dpp(0, s, CTRL, 0xF, 0xF, true);
  return __builtin_bit_cast(float, d);
}
#define QP_ROT1 0x39  // [1,2,3,0]: lane m reads motor (m+1)%4  -> adjacent pair
#define QP_XOR2 0x4E  // [2,3,0,1]: lane m reads motor m^2      -> opposite pair

// One block per batch element. 256 threads = 8 waves (wave32).
// Gram matrix via V_WMMA_F32_16X16X4_F32 with A==B registers => D += A*A^T.
// Rows of A: M = 4*c + m  (c = k-sub-chunk 0..3, m = motor 0..3), so the
// 4 diagonal 4x4 blocks of D are exact per-chunk Gram partials.
__global__ __launch_bounds__(256) void cmfe_kernel(const float* __restrict__ x,
                                                   float* __restrict__ out,
                                                   int T) {
  const int b    = blockIdx.x;
  const int tid  = threadIdx.x;
  const int lane = tid & 31;
  const int wave = tid >> 5;

  const int N = 6 * T;                         // flattened per-motor length
  const float* __restrict__ xb = x + (size_t)b * 4 * (size_t)N;

  // WMMA f32 16x4 A-layout decomposition:
  //   lanes 0-15 hold K=0 (v0) and K=1 (v1); lanes 16-31 hold K=2, K=3.
  const int m  = lane & 3;          // motor (row within chunk block)
  const int c  = (lane >> 2) & 3;   // k-sub-chunk (chunk block)
  const int hi = lane >> 4;         // hi half-wave -> K offset +2
  const float* __restrict__ xm = xb + (size_t)m * (size_t)N;
  const int laneoff = 4 * c + 2 * hi;          // within a 16-k wave tile

  __shared__ float sGram[16];
  __shared__ float sS[4];
  __shared__ float sAdj[4];
  __shared__ float sOpp[2];
  if (tid < 16) sGram[tid] = 0.f;
  if (tid < 4)  { sS[tid] = 0.f; sAdj[tid] = 0.f; }
  if (tid < 2)  sOpp[tid] = 0.f;
  __syncthreads();

  v8f c0 = {}, c1 = {}, c2 = {}, c3 = {};
  float accS = 0.f, accAdj = 0.f, accOpp = 0.f;

  const int rounds = N / 512;                  // 8 waves * 4 tiles * 16 k
  int base = wave * 64 + laneoff;
  for (int t = 0; t < rounds; ++t) {
    const float* p = xm + base;
    // streaming data, no reuse (805MB > 192MB L2): non-temporal loads
    v2f a0 = __builtin_nontemporal_load((const v2f*)(p));
    v2f a1 = __builtin_nontemporal_load((const v2f*)(p + 16));
    v2f a2 = __builtin_nontemporal_load((const v2f*)(p + 32));
    v2f a3 = __builtin_nontemporal_load((const v2f*)(p + 48));

    // D += A*A^T on the matrix pipe (4 independent accumulators)
    c0 = __builtin_amdgcn_wmma_f32_16x16x4_f32(false, a0, false, a0, (short)0, c0, false, false);
    c1 = __builtin_amdgcn_wmma_f32_16x16x4_f32(false, a1, false, a1, (short)0, c1, false, false);
    c2 = __builtin_amdgcn_wmma_f32_16x16x4_f32(false, a2, false, a2, (short)0, c2, false, false);
    c3 = __builtin_amdgcn_wmma_f32_16x16x4_f32(false, a3, false, a3, (short)0, c3, false, false);

    // sums + abs-diffs on VALU (DPP cross-lane) while WMMA pipe drains
#pragma unroll
    for (int u = 0; u < 4; ++u) {
      v2f a = (u == 0) ? a0 : (u == 1) ? a1 : (u == 2) ? a2 : a3;
      float v0 = a.x, v1 = a.y;
      accS += v0 + v1;
      float o0 = quad_perm_f32<QP_XOR2>(v0);   // motor m ^ 2 : opposite pair
      float o1 = quad_perm_f32<QP_XOR2>(v1);
      accOpp += fabsf(v0 - o0) + fabsf(v1 - o1);
      float r0 = quad_perm_f32<QP_ROT1>(v0);   // motor (m+1)%4 : adjacent pair
      float r1 = quad_perm_f32<QP_ROT1>(v1);
      accAdj += fabsf(v0 - r0) + fabsf(v1 - r1);
    }
    base += 512;
  }

  v8f cs = c0 + c1 + c2 + c3;                  // identical layouts -> lanewise add

  // fold diagonal 4x4 chunk-blocks of D into the shared Gram
  const int col = lane & 15;
#pragma unroll
  for (int r = 0; r < 8; ++r) {
    int M = r + (hi ? 8 : 0);
    if ((M >> 2) == (col >> 2))
      atomicAdd(&sGram[(M & 3) * 4 + (col & 3)], cs[r]);
  }
  atomicAdd(&sS[m], accS);
  atomicAdd(&sAdj[m], accAdj);                 // m -> pair (m,(m+1)%4)
  if (m < 2) atomicAdd(&sOpp[m], accOpp);      // lanes m=2,3 are duplicates
  __syncthreads();

  if (tid == 0) {
    const float invN = 1.0f / (float)N;
    float e[4], S[4];
#pragma unroll
    for (int i = 0; i < 4; ++i) { e[i] = sGram[i * 4 + i] * invN; S[i] = sS[i]; }

    float mean = 0.25f * (e[0] + e[1] + e[2] + e[3]);
    float var = 0.f;
#pragma unroll
    for (int i = 0; i < 4; ++i) { float d = e[i] - mean; var += d * d; }
    float estd = sqrtf(var * (1.0f / 3.0f));   // ddof=1
    float emax = fmaxf(fmaxf(e[0], e[1]), fmaxf(e[2], e[3]));
    float emin = fminf(fminf(e[0], e[1]), fminf(e[2], e[3]));

    float Gc[4][4], nrm[4];
#pragma unroll
    for (int i = 0; i < 4; ++i)
#pragma unroll
      for (int j = 0; j < 4; ++j)
        Gc[i][j] = sGram[i * 4 + j] - S[i] * S[j] * invN;
#pragma unroll
    for (int i = 0; i < 4; ++i) nrm[i] = sqrtf(Gc[i][i]);

    float* o = out + (size_t)b * 14;
    o[0] = estd;
    o[1] = emax / (emin + CMFE_EPS);
    const int pi[6] = {0, 0, 0, 1, 1, 2};
    const int pj[6] = {1, 2, 3, 2, 3, 3};
#pragma unroll
    for (int p = 0; p < 6; ++p)
      o[2 + p] = Gc[pi[p]][pj[p]] / (nrm[pi[p]] * nrm[pj[p]] + CMFE_EPS);
    o[8]  = sOpp[0] * invN;   // (0,2)
    o[9]  = sOpp[1] * invN;   // (1,3)
    o[10] = sAdj[0] * invN;   // (0,1)
    o[11] = sAdj[1] * invN;   // (1,2)
    o[12] = sAdj[2] * invN;   // (2,3)
    o[13] = sAdj[3] * invN;   // (3,0)
  }
}

extern "C" void kernel_launch(void* const* d_in, const int* in_sizes, int n_in,
                              void* d_out, int out_size, void* d_ws, size_t ws_size,
                              hipStream_t stream) {
  (void)n_in; (void)d_ws; (void)ws_size;
  const float* x = (const float*)d_in[0];
  float* out = (float*)d_out;
  const int B = out_size / 14;                 // 256
  const int T = in_sizes[0] / (24 * B);        // 32768
  cmfe_kernel<<<B, 256, 0, stream>>>(x, out, T);
}